// MedianFilter1D_66924180406949
// MI455X (gfx1250) — compile-verified
//
#include <hip/hip_runtime.h>

// Median filter 1D, k=9, zero padding. x: [B=16, C=64, L=16384] fp32.
// Bandwidth-bound (128 MiB @ 23.3 TB/s ~ 5.6 us). Stage row tiles into LDS
// via CDNA5 async global->LDS loads (ASYNCcnt), compute median-of-9 with a
// shared-triple med3 network (9.25 three-input VALU ops per output).

#define TILE    2048   // outputs per block
#define PAD     4      // (k-1)/2 halo on each side
#define THREADS 256
#define TPO     8      // outputs per thread

#define AS1 __attribute__((address_space(1)))
#define AS3 __attribute__((address_space(3)))

typedef int v4i __attribute__((vector_size(16)));

#if __has_builtin(__builtin_amdgcn_global_load_async_to_lds_b128) && \
    __has_builtin(__builtin_amdgcn_global_load_async_to_lds_b32)
#define USE_ASYNC 1
#else
#define USE_ASYNC 0
#endif

static __device__ __forceinline__ float min3f(float a, float b, float c) {
  return __builtin_fminf(__builtin_fminf(a, b), c);
}
static __device__ __forceinline__ float max3f(float a, float b, float c) {
  return __builtin_fmaxf(__builtin_fmaxf(a, b), c);
}
static __device__ __forceinline__ float med3f(float a, float b, float c) {
  return __builtin_amdgcn_fmed3f(a, b, c);
}

__global__ __launch_bounds__(THREADS) void MedianFilter1D_66924180406949_kernel(
    const float* __restrict__ x, float* __restrict__ y, int L) {
  __shared__ __align__(16) float tile[TILE + 2 * PAD];

  const int tid        = threadIdx.x;
  const int row        = blockIdx.y;
  const int blockStart = blockIdx.x * TILE;
  const long long rowOff = (long long)row * (long long)L;
  const float* g = x + rowOff;
  float*       o = y + rowOff;

  // ---- Stage [blockStart-4, blockStart+2052) of this row into LDS ----
#if USE_ASYNC
  {
    AS1 v4i* gb4 = (AS1 v4i*)(g + blockStart);
    AS3 v4i* lb4 = (AS3 v4i*)((AS3 float*)tile + PAD);
    // body: 2048 floats = 512 x b128, 2 per thread (always in-row)
    __builtin_amdgcn_global_load_async_to_lds_b128(gb4 + tid, lb4 + tid, 0, 0);
    __builtin_amdgcn_global_load_async_to_lds_b128(gb4 + tid + THREADS,
                                                   lb4 + tid + THREADS, 0, 0);
    // halos: 4 left + 4 right, guarded; zero-pad outside the row
    if (tid < PAD) {
      int gi = blockStart - PAD + tid;
      if (gi >= 0)
        __builtin_amdgcn_global_load_async_to_lds_b32(
            (AS1 int*)(g + gi), (AS3 int*)((AS3 float*)tile + tid), 0, 0);
      else
        tile[tid] = 0.0f;
    } else if (tid < 2 * PAD) {
      int k  = tid - PAD;
      int gi = blockStart + TILE + k;
      int l  = PAD + TILE + k;
      if (gi < L)
        __builtin_amdgcn_global_load_async_to_lds_b32(
            (AS1 int*)(g + gi), (AS3 int*)((AS3 float*)tile + l), 0, 0);
      else
        tile[l] = 0.0f;
    }
#if __has_builtin(__builtin_amdgcn_s_wait_asynccnt)
    __builtin_amdgcn_s_wait_asynccnt(0);
#else
    asm volatile("s_wait_asynccnt 0" ::: "memory");
#endif
  }
#else
  {
    const float4* gb = (const float4*)(g + blockStart);
    float4*       tb = (float4*)(tile + PAD);
    tb[tid]           = gb[tid];
    tb[tid + THREADS] = gb[tid + THREADS];
    if (tid < PAD) {
      int gi    = blockStart - PAD + tid;
      tile[tid] = (gi >= 0) ? g[gi] : 0.0f;
    } else if (tid < 2 * PAD) {
      int k  = tid - PAD;
      int gi = blockStart + TILE + k;
      tile[PAD + TILE + k] = (gi < L) ? g[gi] : 0.0f;
    }
  }
#endif
  __syncthreads();

  // ---- 16-float window slice for this thread's 8 outputs ----
  // Output j = blockStart + 8*tid + t needs x[j-4 .. j+4]  =>  lds[8*tid .. 8*tid+15]
  const float4* t4 = (const float4*)tile + 2 * tid;
  float4 v0 = t4[0], v1 = t4[1], v2 = t4[2], v3 = t4[3];
  float w[16] = {v0.x, v0.y, v0.z, v0.w, v1.x, v1.y, v1.z, v1.w,
                 v2.x, v2.y, v2.z, v2.w, v3.x, v3.y, v3.z, v3.w};

  // Sorted triples (w[t], w[t+1], w[t+2]); shared across neighboring windows.
  float cmn[14], cmd[14], cmx[14];
#pragma unroll
  for (int t = 0; t < 14; ++t) {
    float a = w[t], b = w[t + 1], c = w[t + 2];
    cmn[t] = min3f(a, b, c);
    cmd[t] = med3f(a, b, c);
    cmx[t] = max3f(a, b, c);
  }

  // median9(w[t..t+8]) = med3( max3 of triple-mins, med3 of triple-meds,
  //                            min3 of triple-maxs )
  float r[TPO];
#pragma unroll
  for (int t = 0; t < TPO; ++t) {
    float lo = max3f(cmn[t], cmn[t + 3], cmn[t + 6]);
    float mi = med3f(cmd[t], cmd[t + 3], cmd[t + 6]);
    float hi = min3f(cmx[t], cmx[t + 3], cmx[t + 6]);
    r[t] = med3f(lo, mi, hi);
  }

  float4* op = (float4*)(o + blockStart + TPO * tid);
  op[0] = make_float4(r[0], r[1], r[2], r[3]);
  op[1] = make_float4(r[4], r[5], r[6], r[7]);
}

extern "C" void kernel_launch(void* const* d_in, const int* in_sizes, int n_in,
                              void* d_out, int out_size, void* d_ws, size_t ws_size,
                              hipStream_t stream) {
  (void)n_in; (void)d_ws; (void)ws_size; (void)out_size;
  const float* x = (const float*)d_in[0];
  float*       y = (float*)d_out;

  const int L = 16384;                 // time dimension (reference setup)
  const int R = in_sizes[0] / L;       // B*C rows = 1024
  dim3 grid(L / TILE, R);              // (8, 1024)
  MedianFilter1D_66924180406949_kernel<<<grid, THREADS, 0, stream>>>(x, y, L);
}